// SpatialTransformer_block_4638564679939
// MI455X (gfx1250) — compile-verified
//
#include <hip/hip_runtime.h>
#include <hip/hip_bf16.h>

typedef float v4f __attribute__((ext_vector_type(4)));
typedef float v2f __attribute__((ext_vector_type(2)));

// Problem geometry (fixed by the reference).
constexpr int Dz  = 160;
constexpr int Hy  = 192;
constexpr int Wx  = 224;
constexpr int HW  = Hy * Wx;          // 43008
constexpr int DHW = Dz * HW;          // 6,881,280 voxels (27.5 MB per channel plane)

// One thread handles 4 consecutive x-voxels (W % 4 == 0, rows never split):
// flow reads and output writes are 128-bit; the dx=0/1 corner pair of each
// (dz,dy) corner is fetched with a single b64 gather.
__global__ __launch_bounds__(256) void warp3d_trilinear_kernel(
    const float* __restrict__ src,    // [2, D, H, W]
    const float* __restrict__ flow,   // [3, D, H, W]  ch0=dz, ch1=dy, ch2=dx
    float* __restrict__ out)          // [2, D, H, W]
{
    const int gid  = blockIdx.x * 256 + threadIdx.x;   // voxel-group id
    const int base = gid * 4;
    if (base >= DHW) return;

    // Decompose base -> (z, y, x) of the first of the 4 voxels.
    const int xb = base % Wx;
    const int zy = base / Wx;
    const int yb = zy % Hy;
    const int zb = zy / Hy;

    // Prefetch the streamed flow planes ahead (gfx1250: global_prefetch_b8).
    __builtin_prefetch(flow + base + 16384, 0, 1);
    __builtin_prefetch(flow + DHW + base + 16384, 0, 1);
    __builtin_prefetch(flow + 2 * DHW + base + 16384, 0, 1);

    // Flow is touched exactly once -> non-temporal 128-bit loads; keep L2 for src.
    const v4f fz4 = __builtin_nontemporal_load((const v4f*)(flow + base));
    const v4f fy4 = __builtin_nontemporal_load((const v4f*)(flow + DHW + base));
    const v4f fx4 = __builtin_nontemporal_load((const v4f*)(flow + 2 * DHW + base));

    v4f o0, o1;

    #pragma unroll
    for (int e = 0; e < 4; ++e) {
        // Sample position in index space (align_corners=True semantics).
        const float pz = fz4[e] + (float)zb;
        const float py = fy4[e] + (float)yb;
        const float px = fx4[e] + (float)(xb + e);

        const float zf = floorf(pz), yf = floorf(py), xf = floorf(px);
        const float fz = pz - zf, fy = py - yf, fx = px - xf;
        const int   z0 = (int)zf, y0 = (int)yf, x0 = (int)xf;

        const float wz[2] = {1.0f - fz, fz};
        const float wy[2] = {1.0f - fy, fy};
        const float wx0 = 1.0f - fx, wx1 = fx;

        // x-pair classification: one b64 gather at xp covers both dx corners.
        // Shared across the 4 (dz,dy) pairs and both channels.
        const bool in_mid = ((unsigned)x0 <= (unsigned)(Wx - 2)); // 0..W-2
        const bool is_hi  = (x0 == Wx - 1);   // dx=0 corner valid, value in v[1]
        const bool is_lo  = (x0 == -1);       // dx=1 corner valid, value in v[0]
        const int  xp     = min(max(x0, 0), Wx - 2);  // [xp, xp+1] always in-plane

        float a0 = 0.0f, a1 = 0.0f;

        #pragma unroll
        for (int c = 0; c < 4; ++c) {
            const int dz = (c >> 1) & 1;
            const int dy =  c       & 1;
            const int zi = z0 + dz, yi = y0 + dy;

            const bool inzy = ((unsigned)zi < (unsigned)Dz) &
                              ((unsigned)yi < (unsigned)Hy);
            const int zc = min(max(zi, 0), Dz - 1);
            const int yc = min(max(yi, 0), Hy - 1);
            const int rowbase = (zc * Hy + yc) * Wx + xp;

            float wzy = wz[dz] * wy[dy];
            wzy = inzy ? wzy : 0.0f;              // zeros padding in z/y
            const float w0 = wzy * wx0;           // weight of dx=0 corner
            const float w1 = wzy * wx1;           // weight of dx=1 corner

            // Fold x-edge cases into the pair weights (zeros padding in x).
            const float A = in_mid ? w0 : (is_lo ? w1 : 0.0f);  // applies to v[0]
            const float B = in_mid ? w1 : (is_hi ? w0 : 0.0f);  // applies to v[1]

            // One b64 gather per channel covers both x-corners (dword-aligned).
            v2f p0, p1;
            __builtin_memcpy(&p0, src + rowbase,       8);
            __builtin_memcpy(&p1, src + DHW + rowbase, 8);

            a0 = fmaf(p0.x, A, fmaf(p0.y, B, a0));
            a1 = fmaf(p1.x, A, fmaf(p1.y, B, a1));
        }
        o0[e] = a0;
        o1[e] = a1;
    }

    // Output written once -> non-temporal 128-bit stores.
    __builtin_nontemporal_store(o0, (v4f*)(out + base));
    __builtin_nontemporal_store(o1, (v4f*)(out + DHW + base));
}

extern "C" void kernel_launch(void* const* d_in, const int* in_sizes, int n_in,
                              void* d_out, int out_size, void* d_ws, size_t ws_size,
                              hipStream_t stream) {
    const float* src  = (const float*)d_in[0];  // [1,2,160,192,224] f32
    const float* flow = (const float*)d_in[1];  // [1,3,160,192,224] f32
    float*       out  = (float*)d_out;          // [1,2,160,192,224] f32

    constexpr int groups = DHW / 4;                 // 1,720,320 threads
    constexpr int blocks = (groups + 255) / 256;    // 6720 blocks, exact cover
    warp3d_trilinear_kernel<<<blocks, 256, 0, stream>>>(src, flow, out);
}